// MBNetRoIHead_76124000354842
// MI455X (gfx1250) — compile-verified
//
#include <hip/hip_runtime.h>
#include <hip/hip_bf16.h>
#include <math.h>

// ---------------------------------------------------------------------------
// MBNet RoI head for MI455X (gfx1250).
//   Stage A: per-RoI FPN-level RoIPool(7,7)  -> pooled [256, 256*49] (k = c*49+7h+w)
//   Stage B: h1 = relu(pooled @ conv1_w^T + b1)  (WMMA f32, TDM-fed LDS tiles)
//   Stage C: h2 = relu(h1 @ conv2_w^T + b2)      (same kernel)
//   Stage D: out = [h2@cls_loc_w^T + b | h2@score_w^T + b]  (tiny, VALU)
//
// GEMM1 dominates: 2*256*12544*1024 = 6.6 GFLOP fp32, 51 MB weight stream
// (~2.2 us at 23.3 TB/s). Round-1 codegen serialized load->wait0->wmma; here
// global->LDS movement is done by the Tensor Data Mover (async, TENSORcnt,
// double buffered) so WMMAs only ever wait on short LDS latency.
// ---------------------------------------------------------------------------

typedef float v2f  __attribute__((ext_vector_type(2)));
typedef float v8f  __attribute__((ext_vector_type(8)));
typedef unsigned int u32x4 __attribute__((ext_vector_type(4)));
typedef int   i32x4 __attribute__((ext_vector_type(4)));
typedef int   i32x8 __attribute__((ext_vector_type(8)));

#define R_ROIS   256
#define NCH      256
#define POOL     7
#define KDIM1    (NCH * POOL * POOL)   // 12544
#define HID      1024
#define NCLS_LOC 84
#define NSCORE   21

// --- GEMM tiling ---
#define KB       32                    // K per LDS stage (multiple of 4)
#define ROWF     36                    // padded LDS row stride in floats (32 + 4 pad)
#define WAVES    4                     // waves per block
#define NSPAN    (WAVES * 64)          // 256 output columns per block
#define A_TILE_F (16 * ROWF)           // 576 floats
#define B_TILE_F (NSPAN * ROWF)        // 9216 floats
#define BUF_F    (A_TILE_F + B_TILE_F) // 9792 floats per buffer
#define GEMM_LDS_BYTES (2 * BUF_F * 4) // 78336 bytes (double buffered)

#if __has_builtin(__builtin_amdgcn_tensor_load_to_lds)
#define HAVE_TDM 1
#else
#define HAVE_TDM 0
#endif

__device__ __forceinline__ int iclamp(int v, int lo, int hi) {
    return v < lo ? lo : (v > hi ? hi : v);
}

#if HAVE_TDM
// Issue one TDM 2D tile load: tile_rows x tile_cols fp32 tile at gsrc
// (row pitch = row_stride elements) -> LDS at lds_byte, with D# padding of
// 4 DWORDs per 32 DWORDs => LDS row stride ROWF=36 floats (bank-friendly).
__device__ __forceinline__ void tdm_load_tile(const float* gsrc, unsigned lds_byte,
                                              unsigned rows_rem, unsigned cols_rem,
                                              unsigned tile_rows, unsigned tile_cols,
                                              unsigned row_stride)
{
    unsigned long long ga = ((unsigned long long)(uintptr_t)gsrc) & ((1ULL << 57) - 1ULL);
    u32x4 g0;
    g0.x = 1u;                                        // count=1 (valid), user mode
    g0.y = lds_byte;                                  // D#.lds_addr
    g0.z = (unsigned)(ga & 0xffffffffu);              // global_addr[31:0]
    g0.w = (unsigned)((ga >> 32) & 0x1ffffffu)        // global_addr[56:32]
         | (2u << 30);                                // type = 2 ("image")
    // group1: data_size=4B(2), pad_enable, pad_interval=32DW(code 4),
    // pad_amount=4DW(code 3); no multicast / atomic barrier / iterate.
    const unsigned w0 = (2u << 16) | (1u << 20) | (4u << 22) | (3u << 25);
    const unsigned w1 = (cols_rem & 0xffffu) << 16;                 // tensor_dim0[15:0]
    const unsigned w2 = (cols_rem >> 16) | ((rows_rem & 0xffffu) << 16);
    const unsigned w3 = ((rows_rem >> 16) & 0xffffu) | (tile_cols << 16);
    const unsigned w4 = tile_rows;                                  // tile_dim1 (dim2=0)
    const unsigned w5 = row_stride;                                 // dim0_stride[31:0]
    i32x8 g1 = { (int)w0, (int)w1, (int)w2, (int)w3,
                 (int)w4, (int)w5, 0, 0 };
    i32x4 z4 = { 0, 0, 0, 0 };
#if __has_include(<hip/amd_detail/amd_gfx1250_TDM.h>)
    i32x8 z8 = { 0, 0, 0, 0, 0, 0, 0, 0 };
    __builtin_amdgcn_tensor_load_to_lds(g0, g1, z4, z4, z8, 0);     // 6-arg toolchain
#else
    __builtin_amdgcn_tensor_load_to_lds(g0, g1, z4, z4, 0);         // 5-arg toolchain
#endif
}
#endif  // HAVE_TDM

// One K-block (KB=32) of WMMAs from LDS, software-pipelined one k-step ahead.
// Fragment layout (V_WMMA_F32_16X16X4_F32): lane m/n = lane&15, K-pair
// {k,k+1} for lanes 0-15 and {k+2,k+3} for lanes 16-31 -> ds_load_b64.
__device__ __forceinline__ void kblock_mma(const float* __restrict__ lA,
                                           const float* __restrict__ lB,
                                           v8f* acc)
{
    v2f a  = *(const v2f*)(lA);
    v2f b0 = *(const v2f*)(lB);
    v2f b1 = *(const v2f*)(lB + 16 * ROWF);
    v2f b2 = *(const v2f*)(lB + 32 * ROWF);
    v2f b3 = *(const v2f*)(lB + 48 * ROWF);
    #pragma unroll
    for (int kk = 0; kk < KB; kk += 4) {
        v2f an = {}, b0n = {}, b1n = {}, b2n = {}, b3n = {};
        if (kk + 4 < KB) {               // compile-time after unroll
            an  = *(const v2f*)(lA + kk + 4);
            b0n = *(const v2f*)(lB + kk + 4);
            b1n = *(const v2f*)(lB + 16 * ROWF + kk + 4);
            b2n = *(const v2f*)(lB + 32 * ROWF + kk + 4);
            b3n = *(const v2f*)(lB + 48 * ROWF + kk + 4);
        }
        acc[0] = __builtin_amdgcn_wmma_f32_16x16x4_f32(false, a, false, b0, (short)0, acc[0], false, false);
        acc[1] = __builtin_amdgcn_wmma_f32_16x16x4_f32(false, a, false, b1, (short)0, acc[1], false, false);
        acc[2] = __builtin_amdgcn_wmma_f32_16x16x4_f32(false, a, false, b2, (short)0, acc[2], false, false);
        acc[3] = __builtin_amdgcn_wmma_f32_16x16x4_f32(false, a, false, b3, (short)0, acc[3], false, false);
        a = an; b0 = b0n; b1 = b1n; b2 = b2n; b3 = b3n;
    }
}

// ---------------------------------------------------------------------------
// Stage A: RoI max pooling. One block per RoI, one thread per channel.
// Matches the JAX reference: natural-log level pick, rint quantization,
// MB=6 window cap, empty bins -> 0. Writes pooled with k = c*49 + 7h + w
// so GEMM1 is a plain row-major-K x row-major-K product against conv1_w.
// ---------------------------------------------------------------------------
__global__ void __launch_bounds__(NCH)
roipool_kernel(const float* __restrict__ f2, const float* __restrict__ f3,
               const float* __restrict__ f4, const float* __restrict__ f5,
               const float* __restrict__ rois, const int* __restrict__ bidx,
               float* __restrict__ pooled)
{
    const int r = blockIdx.x;
    const int c = threadIdx.x;

    const float y1 = rois[r * 4 + 0];
    const float x1 = rois[r * 4 + 1];
    const float y2 = rois[r * 4 + 2];
    const float x2 = rois[r * 4 + 3];
    const float hh = y2 - y1 + 1.0f;
    const float ww = x2 - x1 + 1.0f;

    float lf = rintf(logf(sqrtf(hh * ww) / 224.0f) + 4.0f);   // jnp.log = ln
    lf = fminf(fmaxf(lf, 2.0f), 5.0f);
    const int lvl = (int)lf;

    const float* feat;
    int H;
    switch (lvl) {
        case 2:  feat = f2; H = 200; break;
        case 3:  feat = f3; H = 100; break;
        case 4:  feat = f4; H = 50;  break;
        default: feat = f5; H = 25;  break;
    }
    const int W = H;
    const float scale = (float)H / 800.0f;

    const int rsw = (int)rintf(x1 * scale);
    const int rsh = (int)rintf(y1 * scale);
    const int rew = (int)rintf(x2 * scale);
    const int reh = (int)rintf(y2 * scale);
    const int iw = rew - rsw + 1;
    const int ih = reh - rsh + 1;
    const float roi_w = (float)(iw < 1 ? 1 : iw);
    const float roi_h = (float)(ih < 1 ? 1 : ih);
    const float rh7 = roi_h / (float)POOL;
    const float rw7 = roi_w / (float)POOL;

    const int b = bidx[r];
    const float* plane = feat + ((size_t)b * NCH + (size_t)c) * (size_t)H * (size_t)W;
    float* outr = pooled + (size_t)r * KDIM1 + (size_t)c * (POOL * POOL);

    for (int ph = 0; ph < POOL; ++ph) {
        const int hs = iclamp((int)floorf((float)ph * rh7) + rsh, 0, H);
        const int he = iclamp((int)ceilf((float)(ph + 1) * rh7) + rsh, 0, H);
        for (int pw = 0; pw < POOL; ++pw) {
            const int wsb = iclamp((int)floorf((float)pw * rw7) + rsw, 0, W);
            const int web = iclamp((int)ceilf((float)(pw + 1) * rw7) + rsw, 0, W);
            float best;
            if (he <= hs || web <= wsb) {
                best = 0.0f;                       // empty bin
            } else {
                best = -INFINITY;
                #pragma unroll
                for (int dy = 0; dy < 6; ++dy) {   // MB = 6 offset cap
                    const int y = hs + dy;
                    if (y >= he) break;
                    const float* rowp = plane + (size_t)y * W;
                    #pragma unroll
                    for (int dx = 0; dx < 6; ++dx) {
                        const int x = wsb + dx;
                        if (x >= web) break;
                        best = fmaxf(best, rowp[x]);
                    }
                }
            }
            outr[ph * POOL + pw] = best;
        }
    }
}

// ---------------------------------------------------------------------------
// Stages B/C: fp32 WMMA GEMM, TDM-fed double-buffered LDS tiles.
//   D[M,N] = act(A[M,K] @ W[N,K]^T + bias)
// Block = 4 waves; block tile = 16 (M) x 256 (N); wave tile = 16x64.
// Wave 0 drives the TDM; everyone syncs on s_wait_tensorcnt + s_barrier.
// K must be a multiple of KB (12544 and 1024 both are).
// ---------------------------------------------------------------------------
__global__ void __launch_bounds__(WAVES * 32)
wmma_gemm_f32_lds(const float* __restrict__ A, const float* __restrict__ Wm,
                  const float* __restrict__ bias, float* __restrict__ D,
                  int M, int N, int K, int relu)
{
    extern __shared__ float smem[];
    const int tid  = threadIdx.x;
    const int lane = tid & 31;
    const int wv   = tid >> 5;
    const int mTiles = M >> 4;
    const int mT   = blockIdx.x % mTiles;
    const int nB   = blockIdx.x / mTiles;
    const int m0   = mT << 4;
    const int nblk = nB * NSPAN;
    const int row  = lane & 15;
    const int koff = (lane >> 4) << 1;   // 0 or 2

    v8f acc[4] = {v8f{}, v8f{}, v8f{}, v8f{}};
    const int NBK = K / KB;

#if HAVE_TDM
    // Flat LDS pointer low 32 bits == workgroup-relative LDS byte offset.
    const unsigned lds_base = (unsigned)(uintptr_t)smem;
    if (wv == 0) {   // prologue: stage buffer 0
        tdm_load_tile(A  + (size_t)m0   * K, lds_base,                 16,    K, 16,    KB, K);
        tdm_load_tile(Wm + (size_t)nblk * K, lds_base + A_TILE_F * 4,  NSPAN, K, NSPAN, KB, K);
    }
    for (int ib = 0; ib < NBK; ++ib) {
        const int buf = ib & 1;
        if (wv == 0) {
            if (ib + 1 < NBK) {          // prefetch next buffer, then wait current
                const int kb2 = (ib + 1) * KB;
                const unsigned bb = lds_base + (unsigned)((buf ^ 1) * BUF_F * 4);
                tdm_load_tile(A  + (size_t)m0   * K + kb2, bb,                16,    (unsigned)(K - kb2), 16,    KB, K);
                tdm_load_tile(Wm + (size_t)nblk * K + kb2, bb + A_TILE_F * 4, NSPAN, (unsigned)(K - kb2), NSPAN, KB, K);
                __builtin_amdgcn_s_wait_tensorcnt(2);   // current buffer done
            } else {
                __builtin_amdgcn_s_wait_tensorcnt(0);
            }
        }
        __syncthreads();                 // buffer `buf` visible to all waves
        const float* lA = smem + buf * BUF_F + row * ROWF + koff;
        const float* lB = smem + buf * BUF_F + A_TILE_F
                        + (wv * 64 + row) * ROWF + koff;
        kblock_mma(lA, lB, acc);
        __syncthreads();                 // done reading before TDM reuses `buf`
    }
#else
    // Fallback: cooperative synchronous staging, same LDS layout (stride 36).
    for (int ib = 0; ib < NBK; ++ib) {
        const int kb = ib * KB;
        for (int q = tid; q < (16 * 8 + NSPAN * 8); q += WAVES * 32) {
            const float* src;
            float* dst;
            if (q < 16 * 8) {
                const int r4 = q >> 3, c4 = q & 7;
                src = A + (size_t)(m0 + r4) * K + kb + c4 * 4;
                dst = smem + r4 * ROWF + c4 * 4;
            } else {
                const int qq = q - 16 * 8;
                const int r4 = qq >> 3, c4 = qq & 7;
                src = Wm + (size_t)(nblk + r4) * K + kb + c4 * 4;
                dst = smem + A_TILE_F + r4 * ROWF + c4 * 4;
            }
            const float4 v = *(const float4*)src;
            dst[0] = v.x; dst[1] = v.y; dst[2] = v.z; dst[3] = v.w;
        }
        __syncthreads();
        const float* lA = smem + row * ROWF + koff;
        const float* lB = smem + A_TILE_F + (wv * 64 + row) * ROWF + koff;
        kblock_mma(lA, lB, acc);
        __syncthreads();
    }
#endif

    // Store: C/D layout -> VGPR v holds M = v + 8*(lane>=16), N = lane&15.
    const int n0    = nblk + wv * 64;
    const int nbase = n0 + (lane & 15);
    const int mhi   = (lane >> 4) << 3;
    const float bs0 = bias[nbase];
    const float bs1 = bias[nbase + 16];
    const float bs2 = bias[nbase + 32];
    const float bs3 = bias[nbase + 48];
    #pragma unroll
    for (int v = 0; v < 8; ++v) {
        const size_t mrow = (size_t)(m0 + mhi + v) * N;
        float d0 = acc[0][v] + bs0;
        float d1 = acc[1][v] + bs1;
        float d2 = acc[2][v] + bs2;
        float d3 = acc[3][v] + bs3;
        if (relu) {
            d0 = fmaxf(d0, 0.0f); d1 = fmaxf(d1, 0.0f);
            d2 = fmaxf(d2, 0.0f); d3 = fmaxf(d3, 0.0f);
        }
        D[mrow + nbase]      = d0;
        D[mrow + nbase + 16] = d1;
        D[mrow + nbase + 32] = d2;
        D[mrow + nbase + 48] = d3;
    }
}

// ---------------------------------------------------------------------------
// Stage D: two tiny output GEMMs (N=84 and N=21), one thread per output.
// ---------------------------------------------------------------------------
__global__ void __launch_bounds__(256)
head_kernel(const float* __restrict__ h2,
            const float* __restrict__ clw, const float* __restrict__ clb,
            const float* __restrict__ sw,  const float* __restrict__ sb,
            float* __restrict__ out)
{
    const int t = blockIdx.x * blockDim.x + threadIdx.x;
    const int n_cls = R_ROIS * NCLS_LOC;      // 21504
    const int n_tot = n_cls + R_ROIS * NSCORE;
    if (t >= n_tot) return;

    int r, o;
    const float* wrow;
    float acc;
    if (t < n_cls) {
        r = t / NCLS_LOC; o = t - r * NCLS_LOC;
        wrow = clw + (size_t)o * HID;
        acc = clb[o];
    } else {
        const int u = t - n_cls;
        r = u / NSCORE; o = u - r * NSCORE;
        wrow = sw + (size_t)o * HID;
        acc = sb[o];
    }
    const float* h = h2 + (size_t)r * HID;
    #pragma unroll 4
    for (int k = 0; k < HID; k += 4) {
        const float4 hv = *(const float4*)(h + k);
        const float4 wv = *(const float4*)(wrow + k);
        acc = fmaf(hv.x, wv.x, acc);
        acc = fmaf(hv.y, wv.y, acc);
        acc = fmaf(hv.z, wv.z, acc);
        acc = fmaf(hv.w, wv.w, acc);
    }
    out[t] = acc;
}

// ---------------------------------------------------------------------------
// Launch. d_in: feat2..feat5, rois, roi_indices, conv1_w, conv1_b, conv2_w,
// conv2_b, cls_loc_w, cls_loc_b, score_w, score_b, img_h, img_w.
// Workspace: pooled (12.85MB) | h1 (1MB) | h2 (1MB).
// ---------------------------------------------------------------------------
extern "C" void kernel_launch(void* const* d_in, const int* in_sizes, int n_in,
                              void* d_out, int out_size, void* d_ws, size_t ws_size,
                              hipStream_t stream)
{
    const float* f2   = (const float*)d_in[0];
    const float* f3   = (const float*)d_in[1];
    const float* f4   = (const float*)d_in[2];
    const float* f5   = (const float*)d_in[3];
    const float* rois = (const float*)d_in[4];
    const int*   bidx = (const int*)  d_in[5];
    const float* w1   = (const float*)d_in[6];   // [1024, 12544] row-major
    const float* b1   = (const float*)d_in[7];
    const float* w2   = (const float*)d_in[8];   // [1024, 1024]
    const float* b2   = (const float*)d_in[9];
    const float* clw  = (const float*)d_in[10];  // [84, 1024]
    const float* clb  = (const float*)d_in[11];
    const float* sw   = (const float*)d_in[12];  // [21, 1024]
    const float* sb   = (const float*)d_in[13];

    float* pooled = (float*)d_ws;                               // 256*12544 floats
    float* h1     = pooled + (size_t)R_ROIS * KDIM1;            // 256*1024
    float* h2     = h1     + (size_t)R_ROIS * HID;              // 256*1024
    float* out    = (float*)d_out;

    // Stage A
    roipool_kernel<<<dim3(R_ROIS), dim3(NCH), 0, stream>>>(
        f2, f3, f4, f5, rois, bidx, pooled);

    // Stage B: M=256, N=1024, K=12544.
    {
        const int blocks = (R_ROIS / 16) * (HID / NSPAN);       // 16*4 = 64
        wmma_gemm_f32_lds<<<dim3(blocks), dim3(WAVES * 32), GEMM_LDS_BYTES, stream>>>(
            pooled, w1, b1, h1, R_ROIS, HID, KDIM1, 1);
    }

    // Stage C: M=256, N=1024, K=1024.
    {
        const int blocks = (R_ROIS / 16) * (HID / NSPAN);
        wmma_gemm_f32_lds<<<dim3(blocks), dim3(WAVES * 32), GEMM_LDS_BYTES, stream>>>(
            h1, w2, b2, h2, R_ROIS, HID, HID, 1);
    }

    // Stage D: concat [256x84 | 256x21].
    {
        const int n_tot = R_ROIS * (NCLS_LOC + NSCORE);         // 26880
        head_kernel<<<dim3((n_tot + 255) / 256), dim3(256), 0, stream>>>(
            h2, clw, clb, sw, sb, out);
    }
}